// MaskedCTC_76922864271605
// MI455X (gfx1250) — compile-verified
//
#include <hip/hip_runtime.h>

typedef __attribute__((ext_vector_type(2))) float v2f;
typedef __attribute__((ext_vector_type(8))) float v8f;

#define NEGV (-1e30f)
#define ZPAD 84      // LDS stride for 16 x (<=80) z tile
#define AP   520     // LDS stride for 16 x 512 A tile (pad to spread banks)
#define SMAX 1056    // max CTC states (2L+1), L<=64 here -> 129

__device__ __forceinline__ float lae(float a, float b) {
  float mx = fmaxf(a, b);
  float mn = fminf(a, b);
  // both ~NEGV: mx + log1p(1) ~ NEGV (matches jnp.logaddexp semantics here)
  return mx + log1pf(__expf(mn - mx));
}

// ---------------------------------------------------------------------------
// Kernel 1: per-batch candidate list, uniqueness mask, gathered W columns+bias
// cand[b][j]: j=0 -> blank(0), j=1 -> vocab id 1, j=2+l -> labels[l], pad -> 0
// ---------------------------------------------------------------------------
__global__ void ctc_prep(const float* __restrict__ W, const float* __restrict__ bias,
                         const int* __restrict__ ys,
                         int L, int D, int V, int NCPAD,
                         int* __restrict__ cand, float* __restrict__ uniq,
                         float* __restrict__ bc, float* __restrict__ Wc) {
  int b = blockIdx.x;
  __shared__ int sc[128];
  int NC = 2 + L;  // 66
  for (int j = threadIdx.x; j < NCPAD; j += blockDim.x) {
    int c;
    if (j == 0)       c = 0;
    else if (j == 1)  c = 1;
    else if (j < NC)  c = ys[b * L + (j - 2)];
    else              c = 0;
    sc[j] = c;
    cand[b * NCPAD + j] = c;
    bc[b * NCPAD + j]   = bias[c];
  }
  __syncthreads();
  for (int j = threadIdx.x; j < NCPAD; j += blockDim.x) {
    float u = 0.f;
    if (j < NC) {
      u = 1.f;
      int c = sc[j];
      for (int jj = 0; jj < j; ++jj)
        if (sc[jj] == c) { u = 0.f; break; }
    }
    uniq[b * NCPAD + j] = u;
  }
  // gather Wc[b][j][k] = W[k][cand[j]]  (W is D x V row-major)
  int total = NCPAD * D;
  for (int idx = threadIdx.x; idx < total; idx += blockDim.x) {
    int j = idx / D, k = idx - j * D;
    Wc[((size_t)b * NCPAD + j) * D + k] = W[(size_t)k * V + sc[j]];
  }
}

// ---------------------------------------------------------------------------
// Kernel 2: 16(t) x NCPAD(candidate) logits via V_WMMA_F32_16X16X4_F32,
// masked logsumexp over unique candidates, emit lp_ext[b][t][s].
// Block = NT waves (one N-tile of 16 candidates per wave). Grid = (T/16, B).
// ---------------------------------------------------------------------------
__global__ void __launch_bounds__(160)
ctc_gemm_lse(const float* __restrict__ hs, const float* __restrict__ Wc,
             const float* __restrict__ bc, const float* __restrict__ uniq,
             float* __restrict__ lp_ext,
             int T, int D, int L, int NCPAD) {
  int b  = blockIdx.y;
  int t0 = blockIdx.x * 16;
  int S  = 2 * L + 1;

  __shared__ float As[16 * AP];    // A tile: 16 rows x D (D<=512)
  __shared__ float Zs[16 * ZPAD];  // z tile: 16 rows x NCPAD (<=80)
  __shared__ float denom[16];

  // Stage A tile (coalesced float4), reused by all N-tile waves.
  const float* hrow = hs + ((size_t)b * T + t0) * D;
  int nv4 = 16 * (D >> 2);
  for (int idx = threadIdx.x; idx < nv4; idx += blockDim.x) {
    int r = idx / (D >> 2), k4 = idx - r * (D >> 2);
    const float4 v = reinterpret_cast<const float4*>(hrow + (size_t)r * D)[k4];
    float* dst = &As[r * AP + (k4 << 2)];
    dst[0] = v.x; dst[1] = v.y; dst[2] = v.z; dst[3] = v.w;
  }
  __syncthreads();

  int wave  = threadIdx.x >> 5;     // N-tile index
  int lane  = threadIdx.x & 31;
  int j0    = wave * 16;
  int mrow  = lane & 15;            // A: row M; B: column N
  int khalf = (lane >> 4) << 1;     // 0 or 2 (K sub-pair per lane half)

  const float* bcol = Wc + ((size_t)b * NCPAD + j0 + mrow) * D;
  const float* arow = &As[mrow * AP];
  __builtin_prefetch(bcol, 0, 3);   // global_prefetch_b8: warm B column stream

  v8f acc = {};
  for (int k0 = 0; k0 < D; k0 += 4) {
    v2f a, bm;
    a[0]  = arow[k0 + khalf];     a[1]  = arow[k0 + khalf + 1];
    bm[0] = bcol[k0 + khalf];     bm[1] = bcol[k0 + khalf + 1];
    acc = __builtin_amdgcn_wmma_f32_16x16x4_f32(
        /*neg_a=*/false, a, /*neg_b=*/false, bm,
        /*c_mod=*/(short)0, acc, /*reuse_a=*/false, /*reuse_b=*/false);
  }

  // C layout: VGPR r holds (M = r + 8*(lane>=16), N = lane&15)
  int n     = lane & 15;
  int mbase = (lane >> 4) << 3;
  float bj  = bc[b * NCPAD + j0 + n];
  for (int r = 0; r < 8; ++r)
    Zs[(mbase + r) * ZPAD + j0 + n] = acc[r] + bj;
  __syncthreads();

  // Per-row logsumexp over unique candidates (= logsumexp over allowed vocab).
  if (threadIdx.x < 16) {
    int r = threadIdx.x;
    float mx = NEGV;
    for (int j = 0; j < NCPAD; ++j) {
      float z = (uniq[b * NCPAD + j] > 0.f) ? Zs[r * ZPAD + j] : NEGV;
      mx = fmaxf(mx, z);
    }
    float sum = 0.f;
    for (int j = 0; j < NCPAD; ++j)
      if (uniq[b * NCPAD + j] > 0.f) sum += __expf(Zs[r * ZPAD + j] - mx);
    denom[r] = mx + logf(sum);
  }
  __syncthreads();

  // lp_ext[b][t][s]: s even -> blank (j=0); s odd -> label (j = 2 + s/2)
  int tot = 16 * S;
  for (int idx = threadIdx.x; idx < tot; idx += blockDim.x) {
    int r = idx / S, s = idx - r * S;
    int j = (s & 1) ? (2 + (s >> 1)) : 0;
    lp_ext[((size_t)b * T + t0 + r) * S + s] = Zs[r * ZPAD + j] - denom[r];
  }
}

// ---------------------------------------------------------------------------
// Kernel 3: CTC alpha recursion in log space. One block per batch,
// thread-per-state, double-buffered LDS, one barrier per time step.
// ---------------------------------------------------------------------------
__global__ void ctc_forward(const float* __restrict__ lp_ext,
                            const int* __restrict__ ys,
                            const int* __restrict__ hlens,
                            const int* __restrict__ ylens,
                            int T, int L, float* __restrict__ ll) {
  int b = blockIdx.x;
  int S = 2 * L + 1;
  __shared__ float buf[2][SMAX];
  int s = threadIdx.x;
  const float* lpB = lp_ext + (size_t)b * T * S;
  bool active = (s < S);

  bool skip = false;
  if (active && (s & 1) && s >= 3)
    skip = ys[b * L + (s >> 1)] != ys[b * L + ((s - 2) >> 1)];

  if (active)
    buf[0][s] = (s == 0) ? lpB[0] : (s == 1 ? lpB[1] : NEGV);
  __syncthreads();

  int cb = 0;
  int hl = hlens[b];
  for (int t = 1; t < T; ++t) {
    float v = 0.f;
    if (active) {
      float cur = buf[cb][s];
      float a1  = (s >= 1) ? buf[cb][s - 1] : NEGV;
      float a2  = skip ? buf[cb][s - 2] : NEGV;
      v = lae(lae(cur, a1), a2) + lpB[(size_t)t * S + s];
      if (t >= hl) v = cur;      // masked step keeps old alpha
      buf[cb ^ 1][s] = v;        // writes go to the other buffer: no RAW here
    }
    __syncthreads();
    cb ^= 1;
  }
  if (threadIdx.x == 0) {
    int yl = ylens[b];
    ll[b] = lae(buf[cb][2 * yl - 1], buf[cb][2 * yl]);
  }
}

// ---------------------------------------------------------------------------
// Kernel 4: loss = -sum(ll)/B
// ---------------------------------------------------------------------------
__global__ void ctc_reduce(const float* __restrict__ ll, int B,
                           float* __restrict__ out) {
  if (threadIdx.x == 0) {
    float s = 0.f;
    for (int i = 0; i < B; ++i) s += ll[i];
    out[0] = -s / (float)B;
  }
}

// ---------------------------------------------------------------------------
extern "C" void kernel_launch(void* const* d_in, const int* in_sizes, int n_in,
                              void* d_out, int out_size, void* d_ws, size_t ws_size,
                              hipStream_t stream) {
  const float* hs    = (const float*)d_in[0];  // (B,T,D)
  const float* W     = (const float*)d_in[1];  // (D,V)
  const float* bias  = (const float*)d_in[2];  // (V,)
  const int*   hlens = (const int*)d_in[3];    // (B,)
  const int*   ys    = (const int*)d_in[4];    // (B,L)
  const int*   ylens = (const int*)d_in[5];    // (B,)

  int B = in_sizes[3];
  int V = in_sizes[2];
  int D = in_sizes[1] / V;            // 512
  int T = in_sizes[0] / (B * D);      // 1024
  int L = in_sizes[4] / B;            // 64
  int S = 2 * L + 1;                  // 129
  int NC = 2 + L;                     // 66
  int NT = (NC + 15) / 16;            // 5 N-tiles
  int NCPAD = NT * 16;                // 80

  // Workspace carve-up (all 4-byte aligned)
  char* w = (char*)d_ws;
  int*   cand = (int*)w;   w += sizeof(int)   * (size_t)B * NCPAD;
  float* uniq = (float*)w; w += sizeof(float) * (size_t)B * NCPAD;
  float* bc   = (float*)w; w += sizeof(float) * (size_t)B * NCPAD;
  float* ll   = (float*)w; w += sizeof(float) * (size_t)B;
  float* Wc   = (float*)w; w += sizeof(float) * (size_t)B * NCPAD * D;   // 2.6 MB
  float* lp   = (float*)w;                                               // B*T*S ~ 8.5 MB

  ctc_prep<<<B, 256, 0, stream>>>(W, bias, ys, L, D, V, NCPAD, cand, uniq, bc, Wc);

  dim3 gg(T / 16, B);
  ctc_gemm_lse<<<gg, NT * 32, 0, stream>>>(hs, Wc, bc, uniq, lp, T, D, L, NCPAD);

  int thr = ((S + 31) / 32) * 32;     // 160
  ctc_forward<<<B, thr, 0, stream>>>(lp, ys, hlens, ylens, T, L, ll);

  ctc_reduce<<<1, 32, 0, stream>>>(ll, B, (float*)d_out);
}